// DecoderWithAttention_choice_78151224918135
// MI455X (gfx1250) — compile-verified
//
#include <hip/hip_runtime.h>
#include <hip/hip_bf16.h>
#include <math.h>

// ---------------- WMMA helpers (CDNA5 / gfx1250, wave32) ----------------
typedef float v2f __attribute__((ext_vector_type(2)));
typedef float v8f __attribute__((ext_vector_type(8)));

__device__ __forceinline__ v8f wmma_f32_k4(v2f a, v2f b, v8f c) {
  // 8 args: (neg_a, A, neg_b, B, c_mod, C, reuse_a, reuse_b)
  return __builtin_amdgcn_wmma_f32_16x16x4_f32(false, a, false, b, (short)0, c,
                                               false, false);
}

__device__ __forceinline__ float sigmoidf_(float x) {
  return 1.0f / (1.0f + __expf(-x));
}

// Problem constants
#define BB 64
#define LL 51
#define TT 50
#define EE 512
#define HH 512
#define VV 30000
#define ENC 2048
#define G4 2048  // 4*H

// fc tiling: 48 output columns per block, staged through LDS in 2 K-chunks
#define FC_NCOLS 48
#define FC_KCHUNK 256
#define FC_LDS_STRIDE 260  // 256 + 4 pad -> stride mod 64 == 4 (bank spread)

// d_out layout (tuple flattened): predictions | caps | dec_len | sort_ind
#define OUT_PRED 0
#define OUT_CAPS ((size_t)BB * TT * VV)        // 96,000,000
#define OUT_DLEN (OUT_CAPS + (size_t)BB * LL)  // +3264
#define OUT_SIND (OUT_DLEN + (size_t)BB)       // +64

// ---------------- Kernel 1: sort + gather metadata ----------------
__global__ void prep_kernel(const int* __restrict__ caps,
                            const int* __restrict__ cap_len,  // (64,1)
                            float* __restrict__ out,
                            int* __restrict__ sort_ind,
                            int* __restrict__ dec_len,
                            int* __restrict__ caps_s) {
  __shared__ int s_ind[BB];
  if (threadIdx.x == 0) {
    bool taken[BB];
    for (int i = 0; i < BB; ++i) taken[i] = false;
    // stable argsort descending by length (ties: smaller original index first)
    for (int i = 0; i < BB; ++i) {
      int best = -1;
      for (int j = 0; j < BB; ++j) {
        if (!taken[j] && (best < 0 || cap_len[j] > cap_len[best])) best = j;
      }
      taken[best] = true;
      s_ind[i] = best;
    }
  }
  __syncthreads();
  int i = threadIdx.x;
  if (i < BB) {
    int si = s_ind[i];
    sort_ind[i] = si;
    int dl = cap_len[si] - 1;
    dec_len[i] = dl;
    out[OUT_SIND + i] = (float)si;
    out[OUT_DLEN + i] = (float)dl;
    for (int t = 0; t < LL; ++t) {
      int cc = caps[si * LL + t];
      caps_s[i * LL + t] = cc;
      out[OUT_CAPS + (size_t)i * LL + t] = (float)cc;
    }
  }
}

// ---------------- Kernel 2: h0/c0 init GEMM (64x2048)x(2048x512) ----------
// blockDim.x = 128 (4 waves, one per 16-row m-tile); grid (32 n-tiles, 2 h/c)
__global__ void init_kernel(const float* __restrict__ enc,  // 64x2048
                            const int* __restrict__ sort_ind,
                            const float* __restrict__ w_init_h,  // 512x2048
                            const float* __restrict__ b_init_h,
                            const float* __restrict__ w_init_c,
                            const float* __restrict__ b_init_c,
                            float* __restrict__ h,  // 64x512
                            float* __restrict__ c) {
  const int lane = threadIdx.x & 31;
  const int wave = threadIdx.x >> 5;
  const int M0 = wave * 16;
  const int N0 = blockIdx.x * 16;
  const int row16 = lane & 15;
  const int koff = (lane >> 4) * 2;

  const float* W = (blockIdx.y == 0) ? w_init_h : w_init_c;
  const float* bias = (blockIdx.y == 0) ? b_init_h : b_init_c;
  float* dst = (blockIdx.y == 0) ? h : c;

  const float* ap = enc + (size_t)sort_ind[M0 + row16] * ENC + koff;
  const float* bp = W + (size_t)(N0 + row16) * ENC + koff;

  v8f acc = {};
#pragma unroll 4
  for (int k = 0; k < ENC; k += 4) {
    v2f a = *(const v2f*)(ap + k);
    v2f b = *(const v2f*)(bp + k);
    acc = wmma_f32_k4(a, b, acc);
  }

  const int n = N0 + (lane & 15);
  const int mbase = M0 + (lane >> 4) * 8;
  const float bv = bias[n];
#pragma unroll
  for (int r = 0; r < 8; ++r) {
    dst[(mbase + r) * HH + n] = acc[r] + bv;
  }
}

// ---------------- Kernel 3: gates GEMM per step --------------------------
// gates(64x2048) = x_t(64x512) @ w_ih^T + h(64x512) @ w_hh^T
// blockDim.x = 128 (4 waves -> m-tiles); each wave owns a 16x32 tile
// grid 64 (n-super-tiles of 32 columns)
__global__ void gates_kernel(const float* __restrict__ emb_w,  // 30000x512
                             const int* __restrict__ caps_s,   // 64x51
                             const float* __restrict__ h,      // 64x512
                             const float* __restrict__ w_ih,   // 2048x512
                             const float* __restrict__ w_hh,   // 2048x512
                             float* __restrict__ gates,        // 64x2048
                             int t) {
  const int lane = threadIdx.x & 31;
  const int wave = threadIdx.x >> 5;
  const int M0 = wave * 16;
  const int N0 = blockIdx.x * 32;
  const int row16 = lane & 15;
  const int koff = (lane >> 4) * 2;

  const int tok = caps_s[(M0 + row16) * LL + t];
  const float* xap = emb_w + (size_t)tok * EE + koff;
  const float* hap = h + (M0 + row16) * HH + koff;
  const float* bi0 = w_ih + (size_t)(N0 + row16) * EE + koff;
  const float* bi1 = bi0 + (size_t)16 * EE;
  const float* bh0 = w_hh + (size_t)(N0 + row16) * HH + koff;
  const float* bh1 = bh0 + (size_t)16 * HH;

  v8f acc0 = {};
  v8f acc1 = {};
#pragma unroll 4
  for (int k = 0; k < EE; k += 4) {
    v2f a = *(const v2f*)(xap + k);
    v2f b0 = *(const v2f*)(bi0 + k);
    v2f b1 = *(const v2f*)(bi1 + k);
    acc0 = wmma_f32_k4(a, b0, acc0);
    acc1 = wmma_f32_k4(a, b1, acc1);
  }
#pragma unroll 4
  for (int k = 0; k < HH; k += 4) {
    v2f a = *(const v2f*)(hap + k);
    v2f b0 = *(const v2f*)(bh0 + k);
    v2f b1 = *(const v2f*)(bh1 + k);
    acc0 = wmma_f32_k4(a, b0, acc0);
    acc1 = wmma_f32_k4(a, b1, acc1);
  }

  const int n = N0 + (lane & 15);
  const int mbase = M0 + (lane >> 4) * 8;
#pragma unroll
  for (int r = 0; r < 8; ++r) {
    gates[(mbase + r) * G4 + n] = acc0[r];
    gates[(mbase + r) * G4 + n + 16] = acc1[r];
  }
}

// ---------------- Kernel 4: LSTM cell elementwise ------------------------
__global__ void cell_kernel(const float* __restrict__ gates,
                            const float* __restrict__ b_ih,
                            const float* __restrict__ b_hh,
                            float* __restrict__ h, float* __restrict__ c) {
  int idx = blockIdx.x * blockDim.x + threadIdx.x;  // 0..32767
  if (idx >= BB * HH) return;
  int m = idx >> 9;
  int j = idx & (HH - 1);
  const float* g = gates + m * G4;
  float gi = g[j] + b_ih[j] + b_hh[j];
  float gf = g[HH + j] + b_ih[HH + j] + b_hh[HH + j];
  float gg = g[2 * HH + j] + b_ih[2 * HH + j] + b_hh[2 * HH + j];
  float go = g[3 * HH + j] + b_ih[3 * HH + j] + b_hh[3 * HH + j];
  float iv = sigmoidf_(gi);
  float fv = sigmoidf_(gf);
  float gv = tanhf(gg);
  float ov = sigmoidf_(go);
  float cn = fv * c[idx] + iv * gv;
  float hn = ov * tanhf(cn);
  c[idx] = cn;
  h[idx] = hn;
}

// ---------------- Kernel 5: fc GEMM per step (64x512)x(512x30000) --------
// blockDim.x = 128 (4 waves, one per 16-row m-tile); block owns 48 columns.
// B tile (48 rows of w_fc, shared by all 4 waves) is staged in LDS in two
// 256-wide K chunks with coalesced b128 loads; B fragments come from LDS.
// grid = 30000/48 = 625.
__global__ void fc_kernel(const float* __restrict__ h,      // 64x512
                          const float* __restrict__ w_fc,   // 30000x512
                          const float* __restrict__ b_fc,   // 30000
                          const int* __restrict__ dec_len,  // 64
                          float* __restrict__ out, int t) {
  __shared__ float bs[FC_NCOLS * FC_LDS_STRIDE];  // 49,920 bytes

  const int lane = threadIdx.x & 31;
  const int wave = threadIdx.x >> 5;
  const int M0 = wave * 16;
  const int N0 = blockIdx.x * FC_NCOLS;
  const int row16 = lane & 15;
  const int koff = (lane >> 4) * 2;

  const float* ap = h + (M0 + row16) * HH + koff;

  v8f acc0 = {};
  v8f acc1 = {};
  v8f acc2 = {};

  for (int kbase = 0; kbase < HH; kbase += FC_KCHUNK) {
    // ---- cooperative, coalesced stage of B chunk into LDS ----
    for (int i = threadIdx.x; i < FC_NCOLS * (FC_KCHUNK / 4);
         i += blockDim.x) {
      int r = i >> 6;       // 0..47 (FC_KCHUNK/4 == 64 quads per row)
      int cq = i & 63;      // quad-column
      float4 v = *(const float4*)(w_fc + (size_t)(N0 + r) * HH + kbase +
                                  cq * 4);
      *(float4*)(&bs[r * FC_LDS_STRIDE + cq * 4]) = v;
    }
    if (kbase == 0 && threadIdx.x < FC_NCOLS) {
      // warm the next chunk's lines toward cache (global_prefetch_b8)
      __builtin_prefetch(w_fc + (size_t)(N0 + threadIdx.x) * HH + FC_KCHUNK,
                         0, 1);
    }
    __syncthreads();

    // ---- WMMA over this K chunk; B fragments from LDS ----
    const float* bb = &bs[row16 * FC_LDS_STRIDE + koff];
#pragma unroll 4
    for (int k = 0; k < FC_KCHUNK; k += 4) {
      v2f a = *(const v2f*)(ap + kbase + k);
      v2f b0 = *(const v2f*)(bb + k);
      v2f b1 = *(const v2f*)(bb + 16 * FC_LDS_STRIDE + k);
      v2f b2 = *(const v2f*)(bb + 32 * FC_LDS_STRIDE + k);
      acc0 = wmma_f32_k4(a, b0, acc0);
      acc1 = wmma_f32_k4(a, b1, acc1);
      acc2 = wmma_f32_k4(a, b2, acc2);
    }
    __syncthreads();  // protect LDS reuse for next chunk
  }

  const int n = N0 + (lane & 15);
  const int mbase = M0 + (lane >> 4) * 8;
  const float bv0 = b_fc[n];
  const float bv1 = b_fc[n + 16];
  const float bv2 = b_fc[n + 32];
#pragma unroll
  for (int r = 0; r < 8; ++r) {
    int m = mbase + r;
    bool live = (t < dec_len[m]);
    size_t base = OUT_PRED + ((size_t)m * TT + t) * VV + n;
    out[base] = live ? (acc0[r] + bv0) : 0.0f;
    out[base + 16] = live ? (acc1[r] + bv1) : 0.0f;
    out[base + 32] = live ? (acc2[r] + bv2) : 0.0f;
  }
}

// ---------------- Host-side launch ---------------------------------------
extern "C" void kernel_launch(void* const* d_in, const int* in_sizes, int n_in,
                              void* d_out, int out_size, void* d_ws,
                              size_t ws_size, hipStream_t stream) {
  const float* encoder_out = (const float*)d_in[0];
  const int* encoded_captions = (const int*)d_in[1];
  const int* caption_lengths = (const int*)d_in[2];
  const float* embedding_weight = (const float*)d_in[3];
  const float* w_ih = (const float*)d_in[4];
  const float* b_ih = (const float*)d_in[5];
  const float* w_hh = (const float*)d_in[6];
  const float* b_hh = (const float*)d_in[7];
  const float* w_init_h = (const float*)d_in[8];
  const float* b_init_h = (const float*)d_in[9];
  const float* w_init_c = (const float*)d_in[10];
  const float* b_init_c = (const float*)d_in[11];
  const float* w_fc = (const float*)d_in[12];
  const float* b_fc = (const float*)d_in[13];
  float* out = (float*)d_out;

  // workspace layout
  int* sort_ind = (int*)d_ws;             // 64
  int* dec_len = sort_ind + BB;           // 64
  int* caps_s = dec_len + BB;             // 64*51
  float* h = (float*)(caps_s + BB * LL);  // 64*512
  float* c = h + BB * HH;                 // 64*512
  float* gates = c + BB * HH;             // 64*2048

  prep_kernel<<<1, 64, 0, stream>>>(encoded_captions, caption_lengths, out,
                                    sort_ind, dec_len, caps_s);

  init_kernel<<<dim3(HH / 16, 2), 128, 0, stream>>>(
      encoder_out, sort_ind, w_init_h, b_init_h, w_init_c, b_init_c, h, c);

  for (int t = 0; t < TT; ++t) {
    gates_kernel<<<G4 / 32, 128, 0, stream>>>(embedding_weight, caps_s, h,
                                              w_ih, w_hh, gates, t);
    cell_kernel<<<(BB * HH + 255) / 256, 256, 0, stream>>>(gates, b_ih, b_hh,
                                                           h, c);
    fc_kernel<<<VV / FC_NCOLS, 128, 0, stream>>>(h, w_fc, b_fc, dec_len, out,
                                                 t);
  }
}